// GraphNeuralNetwork_30880814858779
// MI455X (gfx1250) — compile-verified
//
#include <hip/hip_runtime.h>
#include <hip/hip_bf16.h>
#include <math.h>

// ---------------- problem constants (match reference) ----------------
constexpr int Bc   = 8;
constexpr int Nn   = 1024;
constexpr int FIN  = 128;
constexpr int Hh   = 128;
constexpr int Ll   = 3;
constexpr int NCLS = 8;
constexpr int Hhalf = Hh / 2; // 64

typedef __attribute__((ext_vector_type(2))) float v2f;
typedef __attribute__((ext_vector_type(8))) float v8f;

// D = A(16x4,f32) * B(4x16,f32) + C(16x16,f32)
// 8-arg form: (neg_a, A, neg_b, B, c_mod, C, reuse_a, reuse_b)
__device__ __forceinline__ v8f wmma_f32(v2f a, v2f b, v8f c) {
    return __builtin_amdgcn_wmma_f32_16x16x4_f32(false, a, false, b, (short)0, c, false, false);
}

__device__ __forceinline__ float sigmoidf(float x) { return 1.0f / (1.0f + expf(-x)); }

// =====================================================================
// Kernel 1: h = emb[types] + nf @ proj_w + proj_b        (M=8192,K=128,N=128)
// one wave per 16x16 output tile
// =====================================================================
__global__ __launch_bounds__(32)
void k_embed(const float* __restrict__ nf, const int* __restrict__ types,
             const float* __restrict__ emb, const float* __restrict__ pw,
             const float* __restrict__ pb, float* __restrict__ h)
{
    const int lane = threadIdx.x;
    const int m    = lane & 15;
    const int half = lane >> 4;
    const int row0 = blockIdx.x * 16;   // over B*N = 8192
    const int col0 = blockIdx.y * 16;   // over H = 128

    v8f c = {};
    for (int k0 = 0; k0 < FIN; k0 += 4) {
        const int ka = k0 + 2 * half;
        v2f a, b;
        const float* arow = nf + (size_t)(row0 + m) * FIN + ka;
        a.x = arow[0];
        a.y = arow[1];
        const float* brow = pw + (size_t)ka * Hh + col0 + m;  // pw[k][n]
        b.x = brow[0];
        b.y = brow[Hh];
        c = wmma_f32(a, b, c);
    }
#pragma unroll
    for (int r = 0; r < 8; ++r) {
        const int row = row0 + r + 8 * half;
        const int col = col0 + m;
        const int ty  = types[row];
        h[(size_t)row * Hh + col] = c[r] + emb[(size_t)ty * Hh + col] + pb[col];
    }
}

// =====================================================================
// Kernel 2: t = h @ W + bias                              (M=8192,K=128,N=128)
// =====================================================================
__global__ __launch_bounds__(32)
void k_lin(const float* __restrict__ A, const float* __restrict__ W,
           const float* __restrict__ bias, float* __restrict__ out)
{
    const int lane = threadIdx.x;
    const int m    = lane & 15;
    const int half = lane >> 4;
    const int row0 = blockIdx.x * 16;
    const int col0 = blockIdx.y * 16;

    v8f c = {};
    for (int k0 = 0; k0 < Hh; k0 += 4) {
        const int ka = k0 + 2 * half;
        v2f a, b;
        const float* arow = A + (size_t)(row0 + m) * Hh + ka;
        a.x = arow[0];
        a.y = arow[1];
        const float* brow = W + (size_t)ka * Hh + col0 + m;
        b.x = brow[0];
        b.y = brow[Hh];
        c = wmma_f32(a, b, c);
    }
#pragma unroll
    for (int r = 0; r < 8; ++r) {
        const int row = row0 + r + 8 * half;
        const int col = col0 + m;
        out[(size_t)row * Hh + col] = c[r] + bias[col];
    }
}

// =====================================================================
// Kernel 3: per-row attention scores  s_src = t·a_src, s_dst = t·a_dst
// =====================================================================
__global__ __launch_bounds__(256)
void k_att(const float* __restrict__ t, const float* __restrict__ attw,
           int layer, float* __restrict__ ssrc, float* __restrict__ sdst)
{
    const int row = blockIdx.x * blockDim.x + threadIdx.x;
    if (row >= Bc * Nn) return;
    const float* tr   = t + (size_t)row * Hh;
    const float* asrc = attw + (size_t)layer * 2 * Hh;
    const float* adst = asrc + Hh;
    float s1 = 0.f, s2 = 0.f;
#pragma unroll 4
    for (int k = 0; k < Hh; ++k) {
        const float v = tr[k];
        s1 += v * asrc[k];
        s2 += v * adst[k];
    }
    ssrc[row] = s1;
    sdst[row] = s2;
}

// =====================================================================
// Kernel 4 (the big one, fused):
//   w[i,j]  = sigmoid(s_src[i]+s_dst[j]+att_b) * (adj0[i,j]>0)
//   agg     = w @ t        (per batch: 1024x1024 @ 1024x128)
//   h       = relu(t + agg)
// Workgroup: 256 threads = 8 waves. Owns (batch b, 16-row block) x full H.
// 16x64 tile of w staged in padded LDS once, reused by all 8 waves.
// =====================================================================
__global__ __launch_bounds__(256)
void k_agg(const float* __restrict__ t, const int* __restrict__ adj0,
           const float* __restrict__ ssrc, const float* __restrict__ sdst,
           const float* __restrict__ attb, int layer, float* __restrict__ hout)
{
    __shared__ float ldsA[64][17];   // [j][i], padded to kill bank conflicts

    const int b    = blockIdx.y;        // batch
    const int i0   = blockIdx.x * 16;   // row block within batch
    const int tid  = threadIdx.x;
    const int lane = tid & 31;
    const int wave = tid >> 5;          // n-tile id, 0..7
    const int m    = lane & 15;
    const int half = lane >> 4;
    const int n0   = wave * 16;

    const float ab = attb[layer];
    const float* tb  = t    + (size_t)b * Nn * Hh;
    const float* ssb = ssrc + (size_t)b * Nn;
    const float* sdb = sdst + (size_t)b * Nn;

    const int ii    = tid & 15;   // i within row block (for LDS fill)
    const int jbase = tid >> 4;   // 0..15
    const float si  = ssb[i0 + ii];

    v8f c = {};
    for (int k0 = 0; k0 < Nn; k0 += 64) {
        __syncthreads();
        // ---- cooperative fill of 16x64 tile of w into LDS ----
#pragma unroll
        for (int rep = 0; rep < 4; ++rep) {
            const int j  = jbase + rep * 16;
            const int gj = k0 + j;
            const int a  = adj0[(size_t)(i0 + ii) * Nn + gj];
            float w = 0.f;
            if (a > 0) w = sigmoidf(si + sdb[gj] + ab);
            ldsA[j][ii] = w;
        }
        __syncthreads();
        // ---- 16 WMMA steps over this K-chunk ----
#pragma unroll
        for (int s = 0; s < 16; ++s) {
            const int ka = s * 4 + 2 * half;
            v2f a, bf;
            a.x = ldsA[ka][m];
            a.y = ldsA[ka + 1][m];
            const float* brow = tb + (size_t)(k0 + ka) * Hh + n0 + m;
            bf.x = brow[0];
            bf.y = brow[Hh];
            c = wmma_f32(a, bf, c);
        }
    }
    // ---- fused epilogue: h = relu(t + agg) ----
#pragma unroll
    for (int r = 0; r < 8; ++r) {
        const int row = i0 + r + 8 * half;
        const int col = n0 + m;
        const size_t idx = (size_t)b * Nn * Hh + (size_t)row * Hh + col;
        const float v = tb[(size_t)row * Hh + col] + c[r];
        hout[idx] = v > 0.f ? v : 0.f;
    }
}

// =====================================================================
// Kernel 5: pooling score per node: a = tanh(h@W1+b1)@w2 + b2
// one wave per node; lane covers 2 of the 64 hidden units
// =====================================================================
__global__ __launch_bounds__(32)
void k_pool_score(const float* __restrict__ h, const float* __restrict__ w1,
                  const float* __restrict__ b1, const float* __restrict__ w2,
                  const float* __restrict__ b2, float* __restrict__ a)
{
    const int row  = blockIdx.x;      // 0..8191
    const int lane = threadIdx.x;
    const float* hr = h + (size_t)row * Hh;
    float acc = 0.f;
#pragma unroll
    for (int uu = 0; uu < 2; ++uu) {
        const int u = lane + uu * 32;
        float d = b1[u];
#pragma unroll 4
        for (int k = 0; k < Hh; ++k) d += hr[k] * w1[(size_t)k * Hhalf + u];
        acc += tanhf(d) * w2[u];
    }
#pragma unroll
    for (int off = 16; off > 0; off >>= 1) acc += __shfl_down(acc, off, 32);
    if (lane == 0) a[row] = acc + b2[0];
}

// =====================================================================
// Kernel 6: softmax over N per batch (1024 threads, in-place -> weights)
// =====================================================================
__global__ __launch_bounds__(1024)
void k_softmax(float* __restrict__ a)
{
    __shared__ float red[32];
    __shared__ float bval[2];
    const int b    = blockIdx.x;
    const int n    = threadIdx.x;       // 0..1023
    const int lane = n & 31;
    const int w    = n >> 5;

    const float v = a[(size_t)b * Nn + n];

    // block max
    float mx = v;
#pragma unroll
    for (int off = 16; off > 0; off >>= 1) mx = fmaxf(mx, __shfl_down(mx, off, 32));
    if (lane == 0) red[w] = mx;
    __syncthreads();
    if (n < 32) {
        float tv = red[lane];
#pragma unroll
        for (int off = 16; off > 0; off >>= 1) tv = fmaxf(tv, __shfl_down(tv, off, 32));
        if (lane == 0) bval[0] = tv;
    }
    __syncthreads();
    const float e = expf(v - bval[0]);

    // block sum
    float s = e;
#pragma unroll
    for (int off = 16; off > 0; off >>= 1) s += __shfl_down(s, off, 32);
    __syncthreads();            // red[] reuse safe
    if (lane == 0) red[w] = s;
    __syncthreads();
    if (n < 32) {
        float tv = red[lane];
#pragma unroll
        for (int off = 16; off > 0; off >>= 1) tv += __shfl_down(tv, off, 32);
        if (lane == 0) bval[1] = tv;
    }
    __syncthreads();
    a[(size_t)b * Nn + n] = e / bval[1];
}

// =====================================================================
// Kernel 7: g[b,c] = sum_n aw[b,n] * h[b,n,c]   (128 threads per batch)
// =====================================================================
__global__ __launch_bounds__(128)
void k_gpool(const float* __restrict__ h, const float* __restrict__ aw,
             float* __restrict__ g)
{
    const int b = blockIdx.x;
    const int c = threadIdx.x;   // 0..127
    const float* hb = h  + (size_t)b * Nn * Hh;
    const float* ab = aw + (size_t)b * Nn;
    float acc = 0.f;
    for (int n = 0; n < Nn; ++n) acc += ab[n] * hb[(size_t)n * Hh + c];
    g[(size_t)b * Hh + c] = acc;
}

// =====================================================================
// Kernel 8: scores = relu(g@W1+b1)@W2 + b2    (64 threads per batch)
// =====================================================================
__global__ __launch_bounds__(64)
void k_cls(const float* __restrict__ g, const float* __restrict__ w1,
           const float* __restrict__ b1, const float* __restrict__ w2,
           const float* __restrict__ b2, float* __restrict__ scores)
{
    __shared__ float r[Hhalf];
    const int b = blockIdx.x;
    const int u = threadIdx.x;   // 0..63
    const float* gb = g + (size_t)b * Hh;
    float d = b1[u];
#pragma unroll 4
    for (int k = 0; k < Hh; ++k) d += gb[k] * w1[(size_t)k * Hhalf + u];
    r[u] = fmaxf(d, 0.f);
    __syncthreads();
    if (u < NCLS) {
        float s = b2[u];
#pragma unroll 4
        for (int k = 0; k < Hhalf; ++k) s += r[k] * w2[(size_t)k * NCLS + u];
        scores[(size_t)b * NCLS + u] = s;
    }
}

// =====================================================================
// host-side launcher
// =====================================================================
extern "C" void kernel_launch(void* const* d_in, const int* in_sizes, int n_in,
                              void* d_out, int out_size, void* d_ws, size_t ws_size,
                              hipStream_t stream)
{
    const float* nf      = (const float*)d_in[0];   // [8,1024,128]
    const int*   adj     = (const int*)  d_in[1];   // [8,1024,1024] (batch 0 used)
    const int*   types   = (const int*)  d_in[2];   // [8,1024]
    const float* emb     = (const float*)d_in[3];   // [10,128]
    const float* proj_w  = (const float*)d_in[4];   // [128,128]
    const float* proj_b  = (const float*)d_in[5];   // [128]
    const float* lin_w   = (const float*)d_in[6];   // [3,128,128]
    const float* lin_b   = (const float*)d_in[7];   // [3,128]
    const float* att_w   = (const float*)d_in[8];   // [3,256]
    const float* att_b   = (const float*)d_in[9];   // [3]
    const float* pool_w1 = (const float*)d_in[10];  // [128,64]
    const float* pool_b1 = (const float*)d_in[11];  // [64]
    const float* pool_w2 = (const float*)d_in[12];  // [64,1]
    const float* pool_b2 = (const float*)d_in[13];  // [1]
    const float* cls_w1  = (const float*)d_in[14];  // [128,64]
    const float* cls_b1  = (const float*)d_in[15];  // [64]
    const float* cls_w2  = (const float*)d_in[16];  // [64,8]
    const float* cls_b2  = (const float*)d_in[17];  // [8]

    float* out_scores = (float*)d_out;        // [8,8]
    float* out_g      = out_scores + Bc*NCLS; // [8,128]

    // workspace layout (floats)
    float* h    = (float*)d_ws;                         // 8*1024*128
    float* t    = h    + (size_t)Bc * Nn * Hh;          // 8*1024*128
    float* ssrc = t    + (size_t)Bc * Nn * Hh;          // 8192
    float* sdst = ssrc + (size_t)Bc * Nn;               // 8192
    float* aw   = sdst + (size_t)Bc * Nn;               // 8192
    (void)ws_size; (void)n_in; (void)in_sizes; (void)out_size;

    const int M = Bc * Nn; // 8192

    // 1) node embedding
    k_embed<<<dim3(M / 16, Hh / 16), 32, 0, stream>>>(nf, types, emb, proj_w, proj_b, h);

    // 2) GNN layers
    for (int l = 0; l < Ll; ++l) {
        k_lin<<<dim3(M / 16, Hh / 16), 32, 0, stream>>>(
            h, lin_w + (size_t)l * Hh * Hh, lin_b + (size_t)l * Hh, t);
        k_att<<<dim3(M / 256), 256, 0, stream>>>(t, att_w, l, ssrc, sdst);
        k_agg<<<dim3(Nn / 16, Bc), 256, 0, stream>>>(t, adj, ssrc, sdst, att_b, l, h);
    }

    // 3) attention pooling
    k_pool_score<<<dim3(M), 32, 0, stream>>>(h, pool_w1, pool_b1, pool_w2, pool_b2, aw);
    k_softmax<<<dim3(Bc), 1024, 0, stream>>>(aw);
    k_gpool<<<dim3(Bc), 128, 0, stream>>>(h, aw, out_g);

    // 4) classifier
    k_cls<<<dim3(Bc), 64, 0, stream>>>(out_g, cls_w1, cls_b1, cls_w2, cls_b2, out_scores);
}